// DecoderRNN_82669530513878
// MI455X (gfx1250) — compile-verified
//
#include <hip/hip_runtime.h>
#include <math.h>

#define NN 64
#define TT 25
#define LLn 196
#define VOCABn 32000
#define EMBEDn 512
#define PROJn 1024
#define DECn 1024
#define G4n 4096
#define KX0n (EMBEDn + PROJn) /* 1536 */

typedef __attribute__((ext_vector_type(2))) float v2f;
typedef __attribute__((ext_vector_type(8))) float v8f;

__device__ __forceinline__ float sigm_(float v) { return 1.0f / (1.0f + expf(-v)); }

// ---------------- init ----------------
__global__ void init_state_kernel(float* __restrict__ h, float* __restrict__ c,
                                  float* __restrict__ ctx,
                                  const float* __restrict__ ih,
                                  const float* __restrict__ ic,
                                  const float* __restrict__ ictx) {
  int idx = blockIdx.x * blockDim.x + threadIdx.x;  // < 3*NN*DECn
  int layer = idx / (NN * DECn);
  int u = idx % DECn;
  h[idx] = ih[layer * DECn + u];
  c[idx] = ic[layer * DECn + u];
  if (idx < NN * PROJn) ctx[idx] = ictx[idx % PROJn];
}

__global__ void out0_kernel(float* __restrict__ out) {
  int idx = blockIdx.x * blockDim.x + threadIdx.x;  // < NN*VOCABn
  int n = idx / VOCABn, j = idx % VOCABn;
  out[(size_t)n * TT * VOCABn + j] = (j == 1) ? 10000.0f : 0.0f;
}

// ---------------- embed + concat(ctx) ----------------
__global__ void embed_concat_kernel(float* __restrict__ x,
                                    const float* __restrict__ emb_table,
                                    const int* __restrict__ captions,
                                    const float* __restrict__ ctx, int s) {
  int idx = blockIdx.x * blockDim.x + threadIdx.x;  // < NN*KX0n
  int n = idx / KX0n, k = idx % KX0n;
  int tok = (s == 0) ? 1 : captions[n * TT + s];
  float v = (k < EMBEDn) ? emb_table[(size_t)tok * EMBEDn + k]
                         : ctx[n * PROJn + (k - EMBEDn)];
  x[idx] = v;
}

// ---------------- fp32 WMMA GEMM (unfused M): out = X*W1^T (+ H*W2^T) + bias ----------------
// one wave per 16x16 tile; nt = wave % Ntiles, mt = wave / Ntiles (mt in 0..3)
__global__ void wmma_gemm_kernel(const float* __restrict__ X, int KX,
                                 const float* __restrict__ W1,
                                 const float* __restrict__ H,
                                 const float* __restrict__ W2, int KH,
                                 const float* __restrict__ b1,
                                 const float* __restrict__ b2,
                                 float* __restrict__ out, unsigned long long ldOut,
                                 int Ntiles) {
  int wave = (blockIdx.x * blockDim.x + threadIdx.x) >> 5;
  int lane = threadIdx.x & 31;
  int half = lane >> 4;   // 0: K pair {0,1}, 1: K pair {2,3}
  int l16  = lane & 15;
  int nt = wave % Ntiles;
  int mt = wave / Ntiles;
  int m0 = mt * 16, n0 = nt * 16;

  // A lane pair: X[m0+l16][k0 + 2*half + {0,1}]  (ISA 16x4 f32 A layout)
  // B lane pair: W^T[k0+2*half+{0,1}][n0+l16] == W1[n0+l16][k0+2*half+{0,1}]
  const float* arow = X  + (size_t)(m0 + l16) * KX + 2 * half;
  const float* brow = W1 + (size_t)(n0 + l16) * KX + 2 * half;

  v8f acc = {};
  for (int k0 = 0; k0 < KX; k0 += 4) {
    if ((k0 & 63) == 0) __builtin_prefetch(brow + k0 + 64, 0, 1);
    v2f a = *(const v2f*)(arow + k0);
    v2f b = *(const v2f*)(brow + k0);
    acc = __builtin_amdgcn_wmma_f32_16x16x4_f32(false, a, false, b,
                                                (short)0, acc, false, false);
  }
  if (H) {  // uniform branch: EXEC stays all-1s
    const float* arow2 = H  + (size_t)(m0 + l16) * KH + 2 * half;
    const float* brow2 = W2 + (size_t)(n0 + l16) * KH + 2 * half;
    for (int k0 = 0; k0 < KH; k0 += 4) {
      if ((k0 & 63) == 0) __builtin_prefetch(brow2 + k0 + 64, 0, 1);
      v2f a = *(const v2f*)(arow2 + k0);
      v2f b = *(const v2f*)(brow2 + k0);
      acc = __builtin_amdgcn_wmma_f32_16x16x4_f32(false, a, false, b,
                                                  (short)0, acc, false, false);
    }
  }

  int col = n0 + l16;
  float bias = b1[col] + (b2 ? b2[col] : 0.0f);
  // C/D layout: lanes0-15: VGPR v = row m0+v; lanes16-31: row m0+8+v; N = lane&15
#pragma unroll
  for (int v = 0; v < 8; ++v) {
    int m = m0 + v + 8 * half;
    out[(size_t)m * ldOut + col] = acc[v] + bias;
  }
}

// ---------------- fp32 WMMA GEMM (fused M=64): one wave owns a 64x16 column strip ----------
// Loads each weight float2 ONCE and feeds 4 WMMAs -> 4x less L2 read traffic on W.
// Used for the logits projection (out_W is the dominant 131 MB/step stream).
__global__ void wmma_gemm_m64_kernel(const float* __restrict__ X, int K,
                                     const float* __restrict__ W1,
                                     const float* __restrict__ b1,
                                     float* __restrict__ out,
                                     unsigned long long ldOut) {
  int wave = (blockIdx.x * blockDim.x + threadIdx.x) >> 5;  // == N tile
  int lane = threadIdx.x & 31;
  int half = lane >> 4;
  int l16  = lane & 15;
  int n0 = wave * 16;

  const float* brow  = W1 + (size_t)(n0 + l16) * K + 2 * half;
  const float* arow0 = X  + (size_t)(l16 +  0) * K + 2 * half;
  const float* arow1 = X  + (size_t)(l16 + 16) * K + 2 * half;
  const float* arow2 = X  + (size_t)(l16 + 32) * K + 2 * half;
  const float* arow3 = X  + (size_t)(l16 + 48) * K + 2 * half;

  v8f acc0 = {}, acc1 = {}, acc2 = {}, acc3 = {};
  for (int k0 = 0; k0 < K; k0 += 4) {
    if ((k0 & 63) == 0) __builtin_prefetch(brow + k0 + 64, 0, 1);
    v2f b  = *(const v2f*)(brow  + k0);
    v2f a0 = *(const v2f*)(arow0 + k0);
    v2f a1 = *(const v2f*)(arow1 + k0);
    v2f a2 = *(const v2f*)(arow2 + k0);
    v2f a3 = *(const v2f*)(arow3 + k0);
    acc0 = __builtin_amdgcn_wmma_f32_16x16x4_f32(false, a0, false, b, (short)0, acc0, false, false);
    acc1 = __builtin_amdgcn_wmma_f32_16x16x4_f32(false, a1, false, b, (short)0, acc1, false, false);
    acc2 = __builtin_amdgcn_wmma_f32_16x16x4_f32(false, a2, false, b, (short)0, acc2, false, false);
    acc3 = __builtin_amdgcn_wmma_f32_16x16x4_f32(false, a3, false, b, (short)0, acc3, false, false);
  }

  int col = n0 + l16;
  float bias = b1[col];
#pragma unroll
  for (int v = 0; v < 8; ++v) {
    int m = v + 8 * half;
    out[(size_t)(m +  0) * ldOut + col] = acc0[v] + bias;
    out[(size_t)(m + 16) * ldOut + col] = acc1[v] + bias;
    out[(size_t)(m + 32) * ldOut + col] = acc2[v] + bias;
    out[(size_t)(m + 48) * ldOut + col] = acc3[v] + bias;
  }
}

// ---------------- LSTM gate nonlinearity (in-place h,c update) ----------------
__global__ void gates_hc_kernel(const float* __restrict__ G,
                                float* __restrict__ h, float* __restrict__ c) {
  int idx = blockIdx.x * blockDim.x + threadIdx.x;  // < NN*DECn
  int n = idx / DECn, u = idx % DECn;
  const float* g = G + (size_t)n * G4n;
  float gi = g[u], gf = g[DECn + u], gg = g[2 * DECn + u], go = g[3 * DECn + u];
  float cc = sigm_(gf) * c[idx] + sigm_(gi) * tanhf(gg);
  float hh = sigm_(go) * tanhf(cc);
  c[idx] = cc;
  h[idx] = hh;
}

// ---------------- dot attention: one block per batch row ----------------
__global__ void attention_kernel(const float* __restrict__ keys,
                                 const float* __restrict__ values,
                                 const float* __restrict__ hlast,
                                 float* __restrict__ ctx) {
  __shared__ float sh[DECn];
  __shared__ float se[LLn];
  int n = blockIdx.x;
  int tid = threadIdx.x;
  int lane = tid & 31, wave = tid >> 5;

  for (int k = tid; k < DECn; k += 256) sh[k] = hlast[n * DECn + k];
  __syncthreads();

  for (int l = wave; l < LLn; l += 8) {
    const float* kr = keys + ((size_t)n * LLn + l) * DECn;
    float s = 0.f;
    for (int k = lane; k < DECn; k += 32) s += kr[k] * sh[k];
    for (int off = 16; off; off >>= 1) s += __shfl_xor(s, off, 32);
    if (lane == 0) se[l] = s;
  }
  __syncthreads();

  if (wave == 0) {
    float m = -3.0e38f;
    for (int i = lane; i < LLn; i += 32) m = fmaxf(m, se[i]);
    for (int off = 16; off; off >>= 1) m = fmaxf(m, __shfl_xor(m, off, 32));
    float s = 0.f;
    for (int i = lane; i < LLn; i += 32) { float e = expf(se[i] - m); se[i] = e; s += e; }
    for (int off = 16; off; off >>= 1) s += __shfl_xor(s, off, 32);
    float inv = 1.0f / s;
    for (int i = lane; i < LLn; i += 32) se[i] *= inv;
  }
  __syncthreads();

  for (int p = tid; p < PROJn; p += 256) {
    float a = 0.f;
    for (int l = 0; l < LLn; ++l)
      a += se[l] * values[((size_t)n * LLn + l) * PROJn + p];
    ctx[n * PROJn + p] = a;
  }
}

// ---------------- host ----------------
extern "C" void kernel_launch(void* const* d_in, const int* in_sizes, int n_in,
                              void* d_out, int out_size, void* d_ws, size_t ws_size,
                              hipStream_t stream) {
  (void)in_sizes; (void)n_in; (void)out_size; (void)ws_size;
  const float* keys     = (const float*)d_in[0];
  const float* values   = (const float*)d_in[1];
  /* d_in[2] features: unused by reference */
  const int*   captions = (const int*)d_in[3];
  /* d_in[4] length: compile-time TT */
  const float* emb      = (const float*)d_in[5];
  const float* Wih0     = (const float*)d_in[6];
  const float* Whh0     = (const float*)d_in[7];
  const float* bih0     = (const float*)d_in[8];
  const float* bhh0     = (const float*)d_in[9];
  const float* Wih_r    = (const float*)d_in[10]; // [2][4096][1024]
  const float* Whh_r    = (const float*)d_in[11];
  const float* bih_r    = (const float*)d_in[12]; // [2][4096]
  const float* bhh_r    = (const float*)d_in[13];
  const float* outW     = (const float*)d_in[14];
  const float* outb     = (const float*)d_in[15];
  const float* init_h   = (const float*)d_in[16];
  const float* init_c   = (const float*)d_in[17];
  const float* init_ctx = (const float*)d_in[18];
  float* out = (float*)d_out;

  float* ws  = (float*)d_ws;
  float* h   = ws;                       // 3*64*1024
  float* c   = h + 3 * NN * DECn;       // 3*64*1024
  float* ctx = c + 3 * NN * DECn;       // 64*1024
  float* x   = ctx + NN * PROJn;        // 64*1536
  float* G   = x + NN * KX0n;           // 64*4096

  init_state_kernel<<<(3 * NN * DECn) / 256, 256, 0, stream>>>(h, c, ctx, init_h, init_c, init_ctx);
  out0_kernel<<<(NN * VOCABn) / 256, 256, 0, stream>>>(out);

  const unsigned long long ldG = G4n;
  const unsigned long long ldO = (unsigned long long)TT * VOCABn;

  for (int s = 0; s < TT - 1; ++s) {
    embed_concat_kernel<<<(NN * KX0n) / 256, 256, 0, stream>>>(x, emb, captions, ctx, s);

    // layer 0: gates = x@Wih0^T + h0@Whh0^T + b   (Ntiles=256 -> 1024 waves -> 128 blocks)
    wmma_gemm_kernel<<<128, 256, 0, stream>>>(x, KX0n, Wih0,
                                              h + 0 * NN * DECn, Whh0, DECn,
                                              bih0, bhh0, G, ldG, 256);
    gates_hc_kernel<<<(NN * DECn) / 256, 256, 0, stream>>>(G, h + 0 * NN * DECn, c + 0 * NN * DECn);

    // layers 1..2
    for (int l = 0; l < 2; ++l) {
      wmma_gemm_kernel<<<128, 256, 0, stream>>>(h + l * NN * DECn, DECn,
                                                Wih_r + (size_t)l * G4n * DECn,
                                                h + (l + 1) * NN * DECn,
                                                Whh_r + (size_t)l * G4n * DECn, DECn,
                                                bih_r + l * G4n, bhh_r + l * G4n,
                                                G, ldG, 256);
      gates_hc_kernel<<<(NN * DECn) / 256, 256, 0, stream>>>(G, h + (l + 1) * NN * DECn,
                                                             c + (l + 1) * NN * DECn);
    }

    const float* hlast = h + 2 * NN * DECn;
    attention_kernel<<<NN, 256, 0, stream>>>(keys, values, hlast, ctx);

    // logits straight into d_out at timestep s+1
    // fused-M kernel: 2000 N-tiles -> 2000 waves -> 250 blocks; out_W read once per step
    wmma_gemm_m64_kernel<<<250, 256, 0, stream>>>(hlast, DECn, outW, outb,
                                                  out + (size_t)(s + 1) * VOCABn, ldO);
  }
}